// SREXmodel_50895362457922
// MI455X (gfx1250) — compile-verified
//
#include <hip/hip_runtime.h>
#include <hip/hip_bf16.h>
#include <stdint.h>

// Problem constants (fixed by the reference)
#define N_NODES 20000
#define F_IN    128
#define H_HEADS 8
#define C_CH    64
#define HC      512
#define E_EDGES 320000
#define ETOT    (E_EDGES + N_NODES)   // edges + self loops
#define R1_     40
#define R2_     50
#define PAIRS_  (R1_ * R2_)
#define LCOLS   39                    // min(R1,R2)-1 logit columns actually used
#define LTOT_   (PAIRS_ * LCOLS)
#define HEADW_LD 61                   // head_W is (1024, 61) row-major
#define HEADW_NPAD 64                 // padded column count for the head GEMM

typedef __attribute__((ext_vector_type(2))) float v2f;
typedef __attribute__((ext_vector_type(8))) float v8f;

// ---------- monotone float<->uint mapping (order-preserving) ----------
__device__ __forceinline__ unsigned f2mono(float f) {
  unsigned u = __float_as_uint(f);
  return (u & 0x80000000u) ? ~u : (u | 0x80000000u);
}
__device__ __forceinline__ float mono2f(unsigned u) {
  return (u & 0x80000000u) ? __uint_as_float(u & 0x7fffffffu)
                           : __uint_as_float(~u);
}

// ---------- B repack into WMMA fragment order ----------
// Bp[(k/2)*Npad*2 + n*2 + j] = (n < Ncopy) ? B[(k+j)*ldb + n] : 0   (j = 0,1)
// After packing, the B fragment {B[ka][n], B[ka+1][n]} is one aligned b64 load.
__global__ void pack_B(const float* __restrict__ B, float* __restrict__ Bp,
                       int K, int Npad, int ldb, int Ncopy) {
  int i = blockIdx.x * blockDim.x + threadIdx.x;   // over (K/2)*Npad
  int tot = (K >> 1) * Npad;
  if (i >= tot) return;
  int k2 = i / Npad, n = i - k2 * Npad;
  float v0 = 0.f, v1 = 0.f;
  if (n < Ncopy) {
    v0 = B[(size_t)(2 * k2) * ldb + n];
    v1 = B[(size_t)(2 * k2 + 1) * ldb + n];
  }
  Bp[(size_t)i * 2]     = v0;
  Bp[(size_t)i * 2 + 1] = v1;
}

// ---------- WMMA fp32 GEMM: D[M, 0:Nout] = A[M,K] @ B[K, 0:Nout] ----------
// B is pre-packed (pack_B) with Npad columns (multiple of 64). One wave32
// computes a 16(M) x 64(N) strip: 4 x v_wmma_f32_16x16x4_f32 per k-step,
// A fragment reused across the 4 N-subtiles. Hot loop is branch-free.
// Requires M % 16 == 0, K % 4 == 0.
__global__ void wmma_gemm_packed(const float* __restrict__ A,
                                 const float* __restrict__ Bp,
                                 float* __restrict__ D,
                                 int M, int Npad, int K, int lda, int ldd, int Nout) {
  const int gwave  = (int)(((size_t)blockIdx.x * blockDim.x + threadIdx.x) >> 5);
  const int lane   = (int)(threadIdx.x & 31);
  const int nQuads = Npad >> 6;
  const int mTiles = M >> 4;
  if (gwave >= mTiles * nQuads) return;   // wave-uniform: EXEC stays all-1s
  const int mT   = gwave / nQuads;
  const int nQ   = gwave - mT * nQuads;
  const int half = lane >> 4;             // 0: K={0,1}, 1: K={2,3} slice
  const int lm   = lane & 15;             // row of A frag / col of B frag
  const int n0   = nQ * 64 + lm;

  const float* Ap = A + (size_t)(mT * 16 + lm) * lda + half * 2;
  const float* Bk = Bp + ((size_t)half * Npad + n0) * 2;
  const size_t bStep = (size_t)2 * Npad * 2;   // k advances 4 -> k/2 advances 2 rows

  v8f acc[4] = {};
#pragma unroll 4
  for (int k = 0; k < K; k += 4) {
    v2f a = *(const v2f*)Ap;
    Ap += 4;
#pragma unroll
    for (int j = 0; j < 4; ++j) {
      v2f b = *(const v2f*)(Bk + j * 32);   // +16 columns => +32 floats
      acc[j] = __builtin_amdgcn_wmma_f32_16x16x4_f32(
          false, a, false, b, (short)0, acc[j], false, false);
    }
    Bk += bStep;
  }

#pragma unroll
  for (int j = 0; j < 4; ++j) {
    const int n = n0 + j * 16;
    if (n < Nout) {
#pragma unroll
      for (int r = 0; r < 8; ++r) {
        D[(size_t)(mT * 16 + r + half * 8) * ldd + n] = acc[j][r];
      }
    }
  }
}

// ---------- small utility kernels ----------
__global__ void fill_f32(float* p, float v, int n) {
  int i = blockIdx.x * blockDim.x + threadIdx.x;
  if (i < n) p[i] = v;
}
__global__ void fill_u32(unsigned* p, unsigned v, int n) {
  int i = blockIdx.x * blockDim.x + threadIdx.x;
  if (i < n) p[i] = v;
}
__global__ void init_bias(float* out, const float* __restrict__ bias, int n) {
  int i = blockIdx.x * blockDim.x + threadIdx.x;
  if (i < n) out[i] = bias[i & (HC - 1)];
}

// a_src[n,h] = dot(h[n,h,:], att_src[h,:]) ; same for dst
__global__ void attn_dots(const float* __restrict__ h,
                          const float* __restrict__ att_src,
                          const float* __restrict__ att_dst,
                          float* __restrict__ a_src, float* __restrict__ a_dst,
                          int n) {
  int i = blockIdx.x * blockDim.x + threadIdx.x;
  if (i >= n) return;
  int node = i >> 3, hd = i & 7;
  const float* hv = h + (size_t)node * HC + hd * C_CH;
  const float* ws = att_src + hd * C_CH;
  const float* wd = att_dst + hd * C_CH;
  float ss = 0.f, sd = 0.f;
#pragma unroll 8
  for (int c = 0; c < C_CH; ++c) {
    float v = hv[c];
    ss += v * ws[c];
    sd += v * wd[c];
  }
  a_src[i] = ss;
  a_dst[i] = sd;
}

// e = leaky_relu(a_src[src] + a_dst[dst]); segment max over dst via mono atomicMax
__global__ void edge_logits(const int* __restrict__ srcp, const int* __restrict__ dstp,
                            const float* __restrict__ a_src, const float* __restrict__ a_dst,
                            float* __restrict__ e, unsigned* __restrict__ mmono, int n) {
  int i = blockIdx.x * blockDim.x + threadIdx.x;
  if (i >= n) return;
  int eid = i >> 3, hd = i & 7;
  int s, d;
  if (eid < E_EDGES) { s = srcp[eid]; d = dstp[eid]; }
  else               { s = d = eid - E_EDGES; }        // self loop
  float x = a_src[s * H_HEADS + hd] + a_dst[d * H_HEADS + hd];
  x = (x > 0.f) ? x : 0.2f * x;
  e[i] = x;
  atomicMax(&mmono[d * H_HEADS + hd], f2mono(x));
}

// e <- exp(e - m[dst]); segment sum over dst
__global__ void edge_exp(const int* __restrict__ dstp,
                         const unsigned* __restrict__ mmono,
                         float* __restrict__ e, float* __restrict__ ssum, int n) {
  int i = blockIdx.x * blockDim.x + threadIdx.x;
  if (i >= n) return;
  int eid = i >> 3, hd = i & 7;
  int d = (eid < E_EDGES) ? dstp[eid] : (eid - E_EDGES);
  float ex = __expf(e[i] - mono2f(mmono[d * H_HEADS + hd]));
  e[i] = ex;
  atomicAdd(&ssum[d * H_HEADS + hd], ex);
}

// out[dst,:] += alpha * h[src,:]   (one block per edge, 256 threads cover 512 ch)
__global__ void edge_scatter(const int* __restrict__ srcp, const int* __restrict__ dstp,
                             const float* __restrict__ e, const float* __restrict__ ssum,
                             const float* __restrict__ h, float* __restrict__ out) {
  int eid = blockIdx.x;
  int s, d;
  if (eid < E_EDGES) { s = srcp[eid]; d = dstp[eid]; }
  else               { s = d = eid - E_EDGES; }
  const float* hs = h + (size_t)s * HC;
  float* od = out + (size_t)d * HC;
#pragma unroll
  for (int c = threadIdx.x; c < HC; c += 256) {
    int hd = c >> 6;
    float alpha = e[(size_t)eid * H_HEADS + hd] / (ssum[d * H_HEADS + hd] + 1e-16f);
    atomicAdd(&od[c], alpha * hs[c]);
  }
}

// agg[route[n], c] += emb[n, c]
__global__ void route_agg(const int* __restrict__ route, const float* __restrict__ emb,
                          float* __restrict__ agg, int n) {
  int i = blockIdx.x * blockDim.x + threadIdx.x;
  if (i >= n) return;
  int node = i >> 9, c = i & (HC - 1);
  atomicAdd(&agg[route[node] * HC + c], emb[i]);
}

// rep[pair, c] = c<512 ? agg1[r1,c] : agg2[r2,c-512]
__global__ void build_rep(const float* __restrict__ agg1, const float* __restrict__ agg2,
                          float* __restrict__ rep, int n) {
  int i = blockIdx.x * blockDim.x + threadIdx.x;
  if (i >= n) return;
  int pair = i >> 10, c = i & 1023;
  int r1 = pair / R2_, r2 = pair - r1 * R2_;
  rep[i] = (c < HC) ? agg1[r1 * HC + c] : agg2[r2 * HC + (c - HC)];
}

// ---------- global softmax over the 78000 logits (with head bias folded in) ----------
__global__ void reduce_max_k(const float* __restrict__ full, const float* __restrict__ head_b,
                             unsigned* __restrict__ gmax) {
  __shared__ float sm[256];
  int i = blockIdx.x * 256 + threadIdx.x;
  float v = -3.4e38f;
  if (i < LTOT_) v = full[i] + head_b[i % LCOLS];
  sm[threadIdx.x] = v;
  __syncthreads();
  for (int s = 128; s > 0; s >>= 1) {
    if ((int)threadIdx.x < s) sm[threadIdx.x] = fmaxf(sm[threadIdx.x], sm[threadIdx.x + s]);
    __syncthreads();
  }
  if (threadIdx.x == 0) atomicMax(gmax, f2mono(sm[0]));
}
__global__ void reduce_sum_k(const float* __restrict__ full, const float* __restrict__ head_b,
                             const unsigned* __restrict__ gmax, float* __restrict__ gsum) {
  __shared__ float sm[256];
  int i = blockIdx.x * 256 + threadIdx.x;
  float mv = mono2f(*gmax);
  float v = 0.f;
  if (i < LTOT_) v = __expf(full[i] + head_b[i % LCOLS] - mv);
  sm[threadIdx.x] = v;
  __syncthreads();
  for (int s = 128; s > 0; s >>= 1) {
    if ((int)threadIdx.x < s) sm[threadIdx.x] += sm[threadIdx.x + s];
    __syncthreads();
  }
  if (threadIdx.x == 0) atomicAdd(gsum, sm[0]);
}
__global__ void write_probs(const float* __restrict__ full, const float* __restrict__ head_b,
                            const unsigned* __restrict__ gmax, const float* __restrict__ gsum,
                            float* __restrict__ probs) {
  int i = blockIdx.x * 256 + threadIdx.x;
  if (i >= LTOT_) return;
  float mv = mono2f(*gmax);
  float inv = 1.0f / (*gsum);
  probs[i] = __expf(full[i] + head_b[i % LCOLS] - mv) * inv;
}

// ---------- host ----------
static inline void launch_gemm(const float* A, const float* Bp, float* D,
                               int M, int Npad, int K, int lda, int ldd, int Nout,
                               hipStream_t s) {
  int nQuads  = Npad >> 6;
  int waves   = (M >> 4) * nQuads;
  int threads = waves * 32;
  int blocks  = (threads + 255) / 256;
  wmma_gemm_packed<<<blocks, 256, 0, s>>>(A, Bp, D, M, Npad, K, lda, ldd, Nout);
}

extern "C" void kernel_launch(void* const* d_in, const int* in_sizes, int n_in,
                              void* d_out, int out_size, void* d_ws, size_t ws_size,
                              hipStream_t stream) {
  const float* x1       = (const float*)d_in[0];
  const int*   ei1      = (const int*)d_in[1];
  const int*   route1   = (const int*)d_in[2];
  const float* x2       = (const float*)d_in[3];
  const int*   ei2      = (const int*)d_in[4];
  const int*   route2   = (const int*)d_in[5];
  const float* lin_W    = (const float*)d_in[8];
  const float* att_src  = (const float*)d_in[9];
  const float* att_dst  = (const float*)d_in[10];
  const float* gat_bias = (const float*)d_in[11];
  const float* head_W   = (const float*)d_in[12];
  const float* head_b   = (const float*)d_in[13];
  float* probs = (float*)d_out;

  char* base = (char*)d_ws;
  size_t off = 0;
  auto alloc = [&](size_t bytes) -> void* {
    void* p = base + off;
    off += (bytes + 255) & ~(size_t)255;
    return p;
  };
  float*    hbuf   = (float*)alloc((size_t)N_NODES * HC * 4);      // reused by both graphs
  float*    out1   = (float*)alloc((size_t)N_NODES * HC * 4);
  float*    out2   = (float*)alloc((size_t)N_NODES * HC * 4);
  float*    asrc   = (float*)alloc((size_t)N_NODES * H_HEADS * 4);
  float*    adst   = (float*)alloc((size_t)N_NODES * H_HEADS * 4);
  unsigned* mmono  = (unsigned*)alloc((size_t)N_NODES * H_HEADS * 4);
  float*    ssum   = (float*)alloc((size_t)N_NODES * H_HEADS * 4);
  float*    ebuf   = (float*)alloc((size_t)ETOT * H_HEADS * 4);
  float*    agg1   = (float*)alloc((size_t)R1_ * HC * 4);
  float*    agg2   = (float*)alloc((size_t)R2_ * HC * 4);
  float*    rep    = (float*)alloc((size_t)PAIRS_ * 2 * HC * 4);
  float*    full   = (float*)alloc((size_t)LTOT_ * 4);
  float*    BpLin  = (float*)alloc((size_t)(F_IN / 2) * HC * 2 * 4);        // 128/2 x 512 x 2
  float*    BpHead = (float*)alloc((size_t)(2 * HC / 2) * HEADW_NPAD * 2 * 4); // 512 x 64 x 2
  unsigned* gmax   = (unsigned*)alloc(4);
  float*    gsum   = (float*)alloc(4);

  // One-time (per call) fragment-order repacks of the two B matrices.
  {
    int tot = (F_IN / 2) * HC;
    pack_B<<<(tot + 255) / 256, 256, 0, stream>>>(lin_W, BpLin, F_IN, HC, HC, HC);
  }
  {
    int tot = (2 * HC / 2) * HEADW_NPAD;
    pack_B<<<(tot + 255) / 256, 256, 0, stream>>>(head_W, BpHead, 2 * HC, HEADW_NPAD,
                                                  HEADW_LD, LCOLS);
  }

  auto gat = [&](const float* x, const int* ei, float* out) {
    launch_gemm(x, BpLin, hbuf, N_NODES, HC, F_IN, F_IN, HC, HC, stream);
    int nh = N_NODES * H_HEADS;
    attn_dots<<<(nh + 255) / 256, 256, 0, stream>>>(hbuf, att_src, att_dst, asrc, adst, nh);
    fill_u32<<<(nh + 255) / 256, 256, 0, stream>>>(mmono, 0u, nh);
    fill_f32<<<(nh + 255) / 256, 256, 0, stream>>>(ssum, 0.f, nh);
    int nhc = N_NODES * HC;
    init_bias<<<(nhc + 255) / 256, 256, 0, stream>>>(out, gat_bias, nhc);
    const int* srcp = ei;
    const int* dstp = ei + E_EDGES;
    int ne = ETOT * H_HEADS;
    edge_logits<<<(ne + 255) / 256, 256, 0, stream>>>(srcp, dstp, asrc, adst, ebuf, mmono, ne);
    edge_exp<<<(ne + 255) / 256, 256, 0, stream>>>(dstp, mmono, ebuf, ssum, ne);
    edge_scatter<<<ETOT, 256, 0, stream>>>(srcp, dstp, ebuf, ssum, hbuf, out);
  };
  gat(x1, ei1, out1);
  gat(x2, ei2, out2);

  fill_f32<<<(R1_ * HC + 255) / 256, 256, 0, stream>>>(agg1, 0.f, R1_ * HC);
  fill_f32<<<(R2_ * HC + 255) / 256, 256, 0, stream>>>(agg2, 0.f, R2_ * HC);
  int nhc = N_NODES * HC;
  route_agg<<<(nhc + 255) / 256, 256, 0, stream>>>(route1, out1, agg1, nhc);
  route_agg<<<(nhc + 255) / 256, 256, 0, stream>>>(route2, out2, agg2, nhc);

  int nrep = PAIRS_ * 2 * HC;
  build_rep<<<(nrep + 255) / 256, 256, 0, stream>>>(agg1, agg2, rep, nrep);

  // full[pair, 0:39] = rep @ head_W[:, 0:39]   (padded to 64 cols, stores guarded)
  launch_gemm(rep, BpHead, full, PAIRS_, HEADW_NPAD, 2 * HC, 2 * HC, LCOLS, LCOLS, stream);

  fill_u32<<<1, 1, 0, stream>>>(gmax, 0u, 1);
  fill_f32<<<1, 1, 0, stream>>>(gsum, 0.f, 1);
  int rb = (LTOT_ + 255) / 256;
  reduce_max_k<<<rb, 256, 0, stream>>>(full, head_b, gmax);
  reduce_sum_k<<<rb, 256, 0, stream>>>(full, head_b, gmax, gsum);
  write_probs<<<rb, 256, 0, stream>>>(full, head_b, gmax, gsum, probs);
}